// AnchorTargetLayer_17300128268726
// MI455X (gfx1250) — compile-verified
//
#include <hip/hip_runtime.h>
#include <hip/hip_bf16.h>
#include <math.h>
#include <stdint.h>

// Problem constants (from reference setup_inputs)
#define Bb   8
#define Aa   200000
#define Nn   100000
#define Gg   64
#define NPOS 128
#define POS_THRf 0.7f
#define NEG_THRf 0.3f
#define EPSf 1.1920929e-07f

#define AS1 __attribute__((address_space(1)))
#define AS3 __attribute__((address_space(3)))

typedef int v4i_t __attribute__((vector_size(16)));  // matches builtin signature

// ---- CDNA5 async global->LDS staging (ASYNCcnt path) ----------------------
__device__ __forceinline__ void async_b128_to_lds(const void* gsrc, void* ldst) {
#if __has_builtin(__builtin_amdgcn_global_load_async_to_lds_b128)
  __builtin_amdgcn_global_load_async_to_lds_b128((AS1 v4i_t*)gsrc, (AS3 v4i_t*)ldst,
                                                 /*imm offset*/0, /*cpol*/0);
#else
  *(float4*)ldst = *(const float4*)gsrc;  // synchronous fallback
#endif
}
__device__ __forceinline__ void wait_async0() {
#if __has_builtin(__builtin_amdgcn_s_wait_asynccnt)
  __builtin_amdgcn_s_wait_asynccnt(0);
#else
  asm volatile("s_wait_asynccnt 0" ::: "memory");
#endif
}

// Deterministic per-element uniform in [0,1)
__device__ __forceinline__ float uhash01(uint32_t x, uint32_t salt) {
  x ^= salt;
  x ^= x >> 16; x *= 0x7feb352dU;
  x ^= x >> 15; x *= 0x846ca68bU;
  x ^= x >> 16;
  return (float)(x >> 8) * (1.0f / 16777216.0f);
}

__device__ __forceinline__ float iou_one(float ax0, float ay0, float ax1, float ay1,
                                         float area_a,
                                         float gx0, float gy0, float gx1, float gy1) {
  float iw    = fminf(ax1, gx1) - fmaxf(ax0, gx0);
  float ih    = fminf(ay1, gy1) - fmaxf(ay0, gy0);
  float inter = fmaxf(iw, 0.0f) * fmaxf(ih, 0.0f);
  float ag    = (gx1 - gx0) * (gy1 - gy0);
  return (inter > 0.0f) ? inter / (area_a + ag - inter) : 0.0f;
}

// ---- K0: initialize outputs + workspace -----------------------------------
__global__ void k_init(int* __restrict__ labels_out, float4* __restrict__ bbox_out,
                       unsigned int* __restrict__ gt_max_bits, int* __restrict__ counts) {
  int idx = blockIdx.x * blockDim.x + threadIdx.x;
  if (idx < Bb * Aa) {
    labels_out[idx] = -1;
    bbox_out[idx]   = make_float4(0.0f, 0.0f, 0.0f, 0.0f);
  }
  if (idx < Bb * Gg) gt_max_bits[idx] = 0u;
  if (idx < 2 * Bb)  counts[idx] = 0;
}

// ---- K1: per-gt global max IoU (gt_max) -----------------------------------
__global__ void k_gtmax(const float* __restrict__ ia,   // (N,4) inside anchors
                        const float* __restrict__ gt,   // (B,G,4)
                        unsigned int* __restrict__ gt_max_bits) {
  __shared__ __align__(16) float sgt[Gg * 4];
  __shared__ unsigned int sgmax[Gg];
  const int b = blockIdx.y;
  const int t = threadIdx.x;
  const int i = blockIdx.x * blockDim.x + t;

  if (t < Gg) {
    async_b128_to_lds(gt + ((size_t)b * Gg + t) * 4, &sgt[t * 4]);
    sgmax[t] = 0u;
  }
  wait_async0();
  __syncthreads();

  // clamp so every lane participates in the wave32 shuffle tree
  const int i2 = (i < Nn) ? i : (Nn - 1);
  if (i + 2048 < Nn) __builtin_prefetch(ia + (size_t)(i + 2048) * 4, 0, 1);
  const float4 a = ((const float4*)ia)[i2];
  const float area_a = (a.z - a.x) * (a.w - a.y);

#pragma unroll 4
  for (int g = 0; g < Gg; ++g) {
    float v = iou_one(a.x, a.y, a.z, a.w, area_a,
                      sgt[g * 4 + 0], sgt[g * 4 + 1], sgt[g * 4 + 2], sgt[g * 4 + 3]);
    // wave32 max-reduction (duplicated tail lanes can't change a max)
    v = fmaxf(v, __shfl_xor(v, 16, 32));
    v = fmaxf(v, __shfl_xor(v, 8, 32));
    v = fmaxf(v, __shfl_xor(v, 4, 32));
    v = fmaxf(v, __shfl_xor(v, 2, 32));
    v = fmaxf(v, __shfl_xor(v, 1, 32));
    if ((t & 31) == 0) atomicMax(&sgmax[g], __float_as_uint(v));  // iou >= 0
  }
  __syncthreads();
  if (t < Gg) atomicMax(&gt_max_bits[b * Gg + t], sgmax[t]);
}

// ---- K2: labels + argmax + per-batch pos/neg counts -----------------------
__global__ void k_label(const float* __restrict__ ia, const float* __restrict__ gt,
                        const unsigned int* __restrict__ gt_max_bits,
                        int* __restrict__ ws_label, int* __restrict__ ws_arg,
                        int* __restrict__ counts) {
  __shared__ __align__(16) float sgt[Gg * 4];
  __shared__ float sgmax[Gg];
  __shared__ int scnt[2];
  const int b = blockIdx.y;
  const int t = threadIdx.x;
  const int i = blockIdx.x * blockDim.x + t;

  if (t < Gg) {
    async_b128_to_lds(gt + ((size_t)b * Gg + t) * 4, &sgt[t * 4]);
    sgmax[t] = __uint_as_float(gt_max_bits[b * Gg + t]);
  }
  if (t < 2) scnt[t] = 0;
  wait_async0();
  __syncthreads();

  if (i < Nn) {
    const float4 a = ((const float4*)ia)[i];
    const float area_a = (a.z - a.x) * (a.w - a.y);
    float amax = 0.0f;
    int   aarg = 0;
    bool  isgt = false;
#pragma unroll 8
    for (int g = 0; g < Gg; ++g) {
      float v = iou_one(a.x, a.y, a.z, a.w, area_a,
                        sgt[g * 4 + 0], sgt[g * 4 + 1], sgt[g * 4 + 2], sgt[g * 4 + 3]);
      if (v > amax) { amax = v; aarg = g; }   // first-max semantics (argmax)
      isgt = isgt || (v == sgmax[g]);
    }
    int lab = (amax >= POS_THRf) ? 1 : ((amax < NEG_THRf) ? 0 : -1);
    if (isgt) lab = 1;
    if (lab == 1) atomicAdd(&scnt[0], 1);
    else if (lab == 0) atomicAdd(&scnt[1], 1);
    ws_label[(size_t)b * Nn + i] = lab;
    ws_arg  [(size_t)b * Nn + i] = aarg;
  }
  __syncthreads();
  if (t == 0) {
    atomicAdd(&counts[b * 2 + 0], scnt[0]);
    atomicAdd(&counts[b * 2 + 1], scnt[1]);
  }
}

// ---- K3: subsample + bbox targets + scatter to full arrays ----------------
__global__ void k_final(const float* __restrict__ ia,
                        const float* __restrict__ ab,            // (A,4) full anchors
                        const int* __restrict__ inside_index,    // (N)
                        const int* __restrict__ ws_label, const int* __restrict__ ws_arg,
                        const int* __restrict__ counts,
                        int* __restrict__ labels_out, float4* __restrict__ bbox_out) {
  __shared__ __align__(16) float sab[Gg * 4];   // reference indexes anchor_boxes[0..63]
  const int b = blockIdx.y;
  const int t = threadIdx.x;
  const int i = blockIdx.x * blockDim.x + t;

  if (t < Gg) async_b128_to_lds(ab + (size_t)t * 4, &sab[t * 4]);
  wait_async0();
  __syncthreads();
  if (i >= Nn) return;

  const int n_pos = counts[b * 2 + 0];
  const int n_neg = counts[b * 2 + 1];
  int lab = ws_label[(size_t)b * Nn + i];
  const uint32_t id = (uint32_t)(b * Nn + i);

  // Deterministic-hash approximation of the reference's random-rank subsampling.
  if (lab == 1 && n_pos > NPOS) {
    if (uhash01(id, 0x9E3779B9u) * (float)n_pos >= (float)NPOS) lab = -1;
  }
  int n_neg2 = n_neg;
  if (n_pos <= NPOS && n_neg > 0) {
    const int need = NPOS - n_pos;
    if (lab == 0 && uhash01(id, 0x85EBCA6Bu) * (float)n_neg < (float)need) lab = 1;
    n_neg2 = n_neg - ((need < n_neg) ? need : n_neg);
  }
  if (lab == 0 && n_neg2 > NPOS) {
    if (uhash01(id, 0xC2B2AE35u) * (float)n_neg2 >= (float)NPOS) lab = -1;
  }

  // bbox regression targets (reproduce reference exactly, incl. cx quirk)
  const float4 a = ((const float4*)ia)[i];
  float w  = a.z - a.x;
  float h  = a.w - a.y;
  const float cx = a.x - 0.5f * w;   // reference: a0 - 0.5*w
  const float cy = a.y + 0.5f * h;   // reference: a1 + 0.5*h
  w = fmaxf(w, EPSf);
  h = fmaxf(h, EPSf);

  const int j = ws_arg[(size_t)b * Nn + i];
  const float m0 = sab[j * 4 + 0], m1 = sab[j * 4 + 1];
  const float m2 = sab[j * 4 + 2], m3 = sab[j * 4 + 3];
  const float bw = m2 - m0, bh = m3 - m1;
  const float bcx = m0 + 0.5f * bw, bcy = m1 + 0.5f * bh;
  const float dx = (bcx - cx) / w;
  const float dy = (bcy - cy) / h;
  const float dw = logf(bw / w);
  const float dh = logf(bh / h);

  const int idx = inside_index[i];
  labels_out[(size_t)b * Aa + idx] = lab;
  bbox_out  [(size_t)b * Aa + idx] = make_float4(dx, dy, dw, dh);
}

// ---------------------------------------------------------------------------
extern "C" void kernel_launch(void* const* d_in, const int* in_sizes, int n_in,
                              void* d_out, int out_size, void* d_ws, size_t ws_size,
                              hipStream_t stream) {
  (void)in_sizes; (void)n_in; (void)out_size; (void)ws_size;
  // inputs per setup_inputs() order:
  //   [0] rpn_cls_score (unused except batch dim), [1] anchor_boxes (A,4),
  //   [2] inside_anchor_boxes (N,4), [3] gt_boxes (B,G,4), [4] inside_index (N)
  const float* ab  = (const float*)d_in[1];
  const float* ia  = (const float*)d_in[2];
  const float* gt  = (const float*)d_in[3];
  const int*   idx = (const int*)d_in[4];

  // outputs: labels_full int32 (B*A) followed by bbox_full float32 (B*A*4)
  int*    labels = (int*)d_out;
  float4* bbox   = (float4*)((float*)d_out + (size_t)Bb * Aa);

  // workspace layout
  uint8_t* ws = (uint8_t*)d_ws;
  unsigned int* gt_max_bits = (unsigned int*)ws;            // B*G*4 = 2 KB
  int* counts   = (int*)(ws + 2048);                        // 2*B ints
  int* ws_label = (int*)(ws + 4096);                        // B*N*4 = 3.2 MB
  int* ws_arg   = (int*)(ws + 4096 + (size_t)Bb * Nn * 4);  // B*N*4 = 3.2 MB

  const dim3 blk(256);
  const dim3 g0((Bb * Aa + 255) / 256);
  const dim3 g1((Nn + 255) / 256, Bb);

  k_init <<<g0, blk, 0, stream>>>(labels, bbox, gt_max_bits, counts);
  k_gtmax<<<g1, blk, 0, stream>>>(ia, gt, gt_max_bits);
  k_label<<<g1, blk, 0, stream>>>(ia, gt, gt_max_bits, ws_label, ws_arg, counts);
  k_final<<<g1, blk, 0, stream>>>(ia, ab, idx, ws_label, ws_arg, counts, labels, bbox);
}